// GraphAttentionV2_63479616635316
// MI455X (gfx1250) — compile-verified
//
#include <hip/hip_runtime.h>
#include <math.h>

typedef float v2f __attribute__((ext_vector_type(2)));
typedef float v8f __attribute__((ext_vector_type(8)));

#define N_NODES 2048
#define BATCH   8
#define FEAT    128
#define NFILT   64

// ---------------------------------------------------------------------------
// Kernel 1: x1[row] = dot(x[row, 0:128], w1)   (row = b*2048+n, 16384 rows)
// One wave32 per row: lane loads float4, wave-reduce via shfl_xor.
// ---------------------------------------------------------------------------
__global__ void k_proj(const float* __restrict__ x, const float* __restrict__ w1,
                       float* __restrict__ X1) {
    int wave = threadIdx.x >> 5;
    int lane = threadIdx.x & 31;
    int row  = blockIdx.x * 8 + wave;                 // 0 .. 16383
    const float4 xv = *(const float4*)(x + (size_t)row * FEAT + lane * 4);
    const float4 wv = *(const float4*)(w1 + lane * 4);
    float d = xv.x * wv.x + xv.y * wv.y + xv.z * wv.z + xv.w * wv.w;
    #pragma unroll
    for (int off = 16; off > 0; off >>= 1) d += __shfl_xor(d, off, 32);
    if (lane == 0) X1[row] = d;
}

// ---------------------------------------------------------------------------
// Kernel 2: WMMA f32 16x16x4.  A-matrix (16 x K) rows 0..7 = x1[b,:],
// row 8 = all ones (produces column-sums cs), rows 9..15 = 0.
// B-matrix = Wa (K x 2048).  One wave per 16-column tile (128 waves).
//   SB2[b][k] = s[b,k] + ba[k]   (rows 0..7 of D)
//   CS[k]     = sum_m Wa[m,k]    (row 8 of D)
// ---------------------------------------------------------------------------
__global__ void k_wmma_s(const float* __restrict__ Wa, const float* __restrict__ ba,
                         const float* __restrict__ X1,
                         float* __restrict__ SB2, float* __restrict__ CS) {
    int wave   = threadIdx.x >> 5;
    int lane   = threadIdx.x & 31;
    int tile   = blockIdx.x * 4 + wave;               // 0..127
    int colIn  = lane & 15;                           // N within tile / A-row M
    int col    = tile * 16 + colIn;                   // Wa column
    int hi     = lane >> 4;                           // 0: K 0,1   1: K 2,3
    int koff   = hi * 2;
    int mrow   = colIn;                               // A-matrix row for this lane
    int srcRow = mrow & 7;

    v8f c = {};
    for (int k0 = 0; k0 < N_NODES; k0 += 4) {
        // A operand (branchless; X1 is 64 KiB -> stays cache-resident)
        v2f av = *(const v2f*)(X1 + srcRow * N_NODES + k0 + koff);
        v2f a;
        a.x = (mrow < 8) ? av.x : ((mrow == 8) ? 1.0f : 0.0f);
        a.y = (mrow < 8) ? av.y : ((mrow == 8) ? 1.0f : 0.0f);
        // B operand: rows k0+koff, k0+koff+1 of Wa at this tile's column
        v2f b;
        b.x = Wa[(size_t)(k0 + koff)     * N_NODES + col];
        b.y = Wa[(size_t)(k0 + koff + 1) * N_NODES + col];
        c = __builtin_amdgcn_wmma_f32_16x16x4_f32(false, a, false, b,
                                                  (short)0, c, false, false);
    }

    float bav = ba[col];
    #pragma unroll
    for (int j = 0; j < 8; ++j) {
        int row = j + hi * 8;                         // D layout: VGPR j -> M=j / j+8
        if (row < 8)        SB2[row * N_NODES + col] = c[j] + bav;
        else if (row == 8)  CS[col] = c[j];
    }
}

// ---------------------------------------------------------------------------
// Kernel 3 (bandwidth-critical): per row (b,n):
//   logit[k] = A[b,n,k] + SB2[b,k] + x1[b,n]*CS[k]
//   out[b,n] = sum_k softmax(logit)[k] * x1[b,k];  write out+bias to 64 filters
// One wave per row; 2048 logits cached in 16 float4 registers (A read once).
// Row-invariant vectors staged in LDS, shared by the block's 8 rows (same b).
// ---------------------------------------------------------------------------
__global__ void __launch_bounds__(256) k_softmax_agg(
        const float* __restrict__ A, const float* __restrict__ bias,
        const float* __restrict__ X1, const float* __restrict__ SB2,
        const float* __restrict__ CS, float* __restrict__ out) {
    __shared__ __align__(16) float css[N_NODES];
    __shared__ __align__(16) float sbs[N_NODES];
    __shared__ __align__(16) float xks[N_NODES];
    __shared__ float bs[NFILT];

    int b     = blockIdx.x >> 8;                      // 256 blocks per batch
    int nBase = (blockIdx.x & 255) * 8;

    for (int i = threadIdx.x; i < N_NODES; i += 256) {
        css[i] = CS[i];
        sbs[i] = SB2[b * N_NODES + i];
        xks[i] = X1[b * N_NODES + i];
    }
    if (threadIdx.x < NFILT) bs[threadIdx.x] = bias[threadIdx.x];
    __syncthreads();

    int wave = threadIdx.x >> 5;
    int lane = threadIdx.x & 31;
    int n    = nBase + wave;
    size_t rowBase = ((size_t)(b * N_NODES + n)) * N_NODES;
    float x1bn = X1[b * N_NODES + n];

    float4 lg[16];
    float  mx = -INFINITY;
    #pragma unroll
    for (int i = 0; i < 16; ++i) {
        int k = i * 128 + lane * 4;                   // coalesced 512B/wave/iter
        float4 av = *(const float4*)(A + rowBase + k);
        float4 cv = *(const float4*)&css[k];
        float4 sv = *(const float4*)&sbs[k];
        float4 l;
        l.x = av.x + sv.x + x1bn * cv.x;
        l.y = av.y + sv.y + x1bn * cv.y;
        l.z = av.z + sv.z + x1bn * cv.z;
        l.w = av.w + sv.w + x1bn * cv.w;
        lg[i] = l;
        mx = fmaxf(mx, fmaxf(fmaxf(l.x, l.y), fmaxf(l.z, l.w)));
    }
    #pragma unroll
    for (int off = 16; off > 0; off >>= 1)
        mx = fmaxf(mx, __shfl_xor(mx, off, 32));

    float Z = 0.0f, W = 0.0f;
    #pragma unroll
    for (int i = 0; i < 16; ++i) {
        int k = i * 128 + lane * 4;
        float4 xv = *(const float4*)&xks[k];
        float ex = __expf(lg[i].x - mx); Z += ex; W += ex * xv.x;
        float ey = __expf(lg[i].y - mx); Z += ey; W += ey * xv.y;
        float ez = __expf(lg[i].z - mx); Z += ez; W += ez * xv.z;
        float ew = __expf(lg[i].w - mx); Z += ew; W += ew * xv.w;
    }
    #pragma unroll
    for (int off = 16; off > 0; off >>= 1) {
        Z += __shfl_xor(Z, off, 32);
        W += __shfl_xor(W, off, 32);
    }
    float r = W / Z;                                  // all lanes hold result

    size_t ob = ((size_t)(b * N_NODES + n)) * NFILT;
    out[ob + lane]      = r + bs[lane];
    out[ob + 32 + lane] = r + bs[32 + lane];
}

// ---------------------------------------------------------------------------
extern "C" void kernel_launch(void* const* d_in, const int* in_sizes, int n_in,
                              void* d_out, int out_size, void* d_ws, size_t ws_size,
                              hipStream_t stream) {
    const float* x    = (const float*)d_in[0];   // [8,2048,128]
    const float* A    = (const float*)d_in[1];   // [8,2048,2048]
    const float* w1   = (const float*)d_in[2];   // [128,1]
    const float* Wa   = (const float*)d_in[3];   // [2048,2048]
    const float* ba   = (const float*)d_in[4];   // [2048]
    const float* bias = (const float*)d_in[5];   // [64]
    float* out = (float*)d_out;                  // [8,2048,64]

    float* X1  = (float*)d_ws;                   // 16384 floats
    float* CS  = X1 + BATCH * N_NODES;           // 2048 floats
    float* SB2 = CS + N_NODES;                   // 16384 floats

    // 16384 rows / 8 waves per block
    k_proj<<<dim3(2048), dim3(256), 0, stream>>>(x, w1, X1);
    // 128 column tiles / 4 waves per block
    k_wmma_s<<<dim3(32), dim3(128), 0, stream>>>(Wa, ba, X1, SB2, CS);
    // 8*2048 rows / 8 rows per block
    k_softmax_agg<<<dim3(2048), dim3(256), 0, stream>>>(A, bias, X1, SB2, CS, out);
}